// VectorQuantizer_23158463660762
// MI455X (gfx1250) — compile-verified
//
#include <hip/hip_runtime.h>

typedef float v2f __attribute__((ext_vector_type(2)));
typedef float v8f __attribute__((ext_vector_type(8)));

#define K_CODES 512
#define DIM     64
#define HW      4096
#define CHW     262144
#define TILE_M  128
#define NUMEL   8388608

// workspace layout (floats)
#define WS_LOSS 0
#define WS_CS   1
#define WS_AVG  513
#define WS_NORM 33281   // 512 code norms

#define B_STRIDE 68     // padded code-tile row stride (272B, 16B aligned)
#define A_STRIDE 132    // padded x-tile row stride (528B, 16B aligned)

__device__ __forceinline__ void atomic_add_f32(float* p, float v) {
  __hip_atomic_fetch_add(p, v, __ATOMIC_RELAXED, __HIP_MEMORY_SCOPE_AGENT);
}

// CDNA5 async global->LDS copy, 16B per lane, GV mode (SADDR=off).
// VDST operand carries the LDS byte offset (low 32 bits of the flat shared
// address per the aperture rule), VADDR the 64-bit global address.
__device__ __forceinline__ void async_copy_b128(void* lds_ptr, const void* gptr) {
  uint32_t loff = (uint32_t)(uintptr_t)lds_ptr;
  asm volatile("global_load_async_to_lds_b128 %0, %1, off"
               :: "v"(loff), "v"(gptr) : "memory");
}
__device__ __forceinline__ void async_wait0() {
  asm volatile("s_wait_asynccnt 0x0" ::: "memory");
}

// Zero the accumulators, precompute ||e_k||^2
__global__ void vq_init(const float* __restrict__ emb, float* __restrict__ ws) {
  int gid = blockIdx.x * blockDim.x + threadIdx.x;
  if (gid < WS_NORM) ws[gid] = 0.0f;
  if (gid < K_CODES) {
    float s = 0.f;
    #pragma unroll 8
    for (int d = 0; d < DIM; ++d) { float e = emb[gid * DIM + d]; s += e * e; }
    ws[WS_NORM + gid] = s;
  }
}

// Main: distance GEMM via f32 WMMA (async double-buffered B, hoisted A
// fragments) + argmin + quantize + loss + segment sums
__launch_bounds__(256)
__global__ void vq_main(const float* __restrict__ x,
                        const float* __restrict__ emb,
                        float* __restrict__ out_q,
                        float* __restrict__ ws) {
  __shared__ float A_lds[DIM * A_STRIDE];      // x tile [d][t]
  __shared__ float B_lds[2][16 * B_STRIDE];    // double-buffered code tile
  __shared__ int   idx_lds[TILE_M];
  __shared__ float red[256];

  const int tid = threadIdx.x;
  const int tile = blockIdx.x;
  const int n = tile >> 5;                  // 32 tiles per image (HW/128)
  const int planeBase = (tile & 31) * TILE_M;
  const float* xblk = x + (size_t)n * CHW + planeBase;

  // my B-staging chunk: 16 codes x 16 16B-chunks = 256 copies, 1/thread
  const int bcode = tid >> 4;
  const int bd4   = (tid & 15) * 4;

  // Prologue: issue A tile (64 rows x 32 chunks) and B tile 0.
  for (int i = tid; i < DIM * (TILE_M / 4); i += 256) {
    int d = i >> 5, t4 = (i & 31) * 4;
    async_copy_b128(&A_lds[d * A_STRIDE + t4], xblk + d * HW + t4);
  }
  async_copy_b128(&B_lds[0][bcode * B_STRIDE + bd4],
                  emb + bcode * DIM + bd4);
  async_wait0();
  __syncthreads();                          // A tile + B tile 0 visible

  const int lane = tid & 31;
  const int wv   = tid >> 5;   // wave id 0..7 -> 16-token row block
  const int lm   = lane & 15;
  const int lh   = lane >> 4;  // half-wave: selects K pair / row half
  const int wrow = wv * 16;

  // Hoist the loop-invariant A fragments (16 k-steps x 2 VGPRs) once.
  v2f af[16];
  #pragma unroll
  for (int kk = 0; kk < 16; ++kk) {
    int k0 = kk * 4 + lh * 2;
    af[kk].x = A_lds[(k0 + 0) * A_STRIDE + wrow + lm];
    af[kk].y = A_lds[(k0 + 1) * A_STRIDE + wrow + lm];
  }

  float bestVal[8];
  int   bestIdx[8];
  #pragma unroll
  for (int r = 0; r < 8; ++r) { bestVal[r] = 3.4e38f; bestIdx[r] = 0; }

  for (int j = 0; j < 32; ++j) {          // 32 N-tiles of 16 codes
    if (j + 1 < 32) {                     // prefetch next code tile
      async_copy_b128(&B_lds[(j + 1) & 1][bcode * B_STRIDE + bd4],
                      emb + ((j + 1) * 16 + bcode) * DIM + bd4);
    }

    const float* Bcur = B_lds[j & 1];
    v8f acc = {0.f, 0.f, 0.f, 0.f, 0.f, 0.f, 0.f, 0.f};
    #pragma unroll
    for (int kk = 0; kk < 16; ++kk) {     // K=64 in steps of 4
      int k0 = kk * 4 + lh * 2;
      v2f b;
      b.x = Bcur[lm * B_STRIDE + k0 + 0];
      b.y = Bcur[lm * B_STRIDE + k0 + 1];
      acc = __builtin_amdgcn_wmma_f32_16x16x4_f32(
          false, af[kk], false, b, (short)0, acc, false, false);
    }

    // score = ||e||^2 - 2*dot   (||x||^2 constant per row -> irrelevant)
    float en  = ws[WS_NORM + j * 16 + lm];
    int   col = j * 16 + lm;
    #pragma unroll
    for (int r = 0; r < 8; ++r) {
      float s = en - 2.0f * acc[r];
      if (s < bestVal[r]) { bestVal[r] = s; bestIdx[r] = col; }
    }

    if (j + 1 < 32) {
      async_wait0();                      // next buffer landed...
      __syncthreads();                    // ...and visible to all waves
    }
  }

  // argmin across the 16 lanes of each half-wave (C layout: one col per lane)
  #pragma unroll
  for (int m = 1; m < 16; m <<= 1) {
    #pragma unroll
    for (int r = 0; r < 8; ++r) {
      float ov = __shfl_xor(bestVal[r], m, 32);
      int   oi = __shfl_xor(bestIdx[r], m, 32);
      if (ov < bestVal[r] || (ov == bestVal[r] && oi < bestIdx[r])) {
        bestVal[r] = ov; bestIdx[r] = oi;
      }
    }
  }
  if (lm == 0) {
    #pragma unroll
    for (int r = 0; r < 8; ++r)
      idx_lds[wrow + lh * 8 + r] = bestIdx[r];
  }
  __syncthreads();

  // quantized output (straight-through == quantized numerically) + loss
  float lsum = 0.f;
  for (int i = tid; i < DIM * TILE_M; i += 256) {
    int c = i >> 7, t = i & 127;
    int k = idx_lds[t];
    float q  = emb[k * DIM + c];
    float xv = A_lds[c * A_STRIDE + t];
    float dd = xv - q;
    lsum += dd * dd;
    out_q[(size_t)n * CHW + c * HW + planeBase + t] = q;
  }
  red[tid] = lsum;
  __syncthreads();
  for (int s = 128; s > 0; s >>= 1) {
    if (tid < s) red[tid] += red[tid + s];
    __syncthreads();
  }
  if (tid == 0) atomic_add_f32(&ws[WS_LOSS], red[0]);

  // segment sums: histogram + feature scatter-add
  if (tid < TILE_M) atomic_add_f32(&ws[WS_CS + idx_lds[tid]], 1.0f);
  for (int i = tid; i < TILE_M * DIM; i += 256) {
    int t = i >> 6, d = i & 63;
    atomic_add_f32(&ws[WS_AVG + idx_lds[t] * DIM + d], A_lds[d * A_STRIDE + t]);
  }
}

// EMA updates + loss mean
__global__ void vq_finalize(const float* __restrict__ cs,
                            const float* __restrict__ eavg,
                            const float* __restrict__ ws,
                            float* __restrict__ out_loss,
                            float* __restrict__ out_emb,
                            float* __restrict__ out_cs,
                            float* __restrict__ out_avg) {
  int gid = blockIdx.x * blockDim.x + threadIdx.x;
  if (gid >= K_CODES * DIM) return;
  int k = gid >> 6, d = gid & 63;
  float ncs  = cs[k]     * 0.99f + 0.01f * ws[WS_CS + k];
  float navg = eavg[gid] * 0.99f + 0.01f * ws[WS_AVG + gid];
  out_avg[gid] = navg;
  out_emb[gid] = navg / (ncs + 1e-5f);
  if (d == 0)   out_cs[k] = ncs;
  if (gid == 0) out_loss[0] = ws[WS_LOSS] / (float)NUMEL;
}

extern "C" void kernel_launch(void* const* d_in, const int* in_sizes, int n_in,
                              void* d_out, int out_size, void* d_ws, size_t ws_size,
                              hipStream_t stream) {
  (void)in_sizes; (void)n_in; (void)out_size; (void)ws_size;
  const float* x    = (const float*)d_in[0];
  const float* emb  = (const float*)d_in[1];
  const float* cs   = (const float*)d_in[2];
  const float* eavg = (const float*)d_in[3];
  float* out = (float*)d_out;
  float* ws  = (float*)d_ws;

  float* out_q    = out;                         // 8388608
  float* out_loss = out + 8388608;               // 1
  float* out_emb  = out + 8388609;               // 32768
  float* out_cs   = out + 8388609 + 32768;       // 512
  float* out_avg  = out + 8388609 + 32768 + 512; // 32768

  vq_init<<<(WS_NORM + 255) / 256, 256, 0, stream>>>(emb, ws);
  vq_main<<<1024, 256, 0, stream>>>(x, emb, out_q, ws);
  vq_finalize<<<(K_CODES * DIM + 255) / 256, 256, 0, stream>>>(
      cs, eavg, ws, out_loss, out_emb, out_cs, out_avg);
}